// GT2RBFnn_23072564314452
// MI455X (gfx1250) — compile-verified
//
#include <hip/hip_runtime.h>

typedef __attribute__((ext_vector_type(2))) float v2f;
typedef __attribute__((ext_vector_type(8))) float v8f;
typedef __attribute__((ext_vector_type(4))) unsigned int u32x4;
typedef __attribute__((ext_vector_type(8))) int i32x8;
typedef __attribute__((ext_vector_type(4))) int i32x4;

#define NN 8
#define MM 16
#define AS 101
#define PP 3
#define LOG2E 1.4426950408889634f

#define TBL_ELEMS (MM * AS * NN)          // 12928 floats per table
#define TBL_BYTES (TBL_ELEMS * 4)         // 51712 bytes
#define LDS_BYTES (2 * TBL_BYTES)         // 103424 bytes (< 320KB/WGP)

// Kernel 1: build per-(m, alpha) coefficient tables with n innermost:
//   c[m][a][n] = -0.5*log2(e) / (sig +/- delta*(1-alpha))^2
__global__ void build_tables(const float* __restrict__ sigma,
                             const float* __restrict__ delta_sig,
                             const float* __restrict__ alpha,
                             float* __restrict__ cl,
                             float* __restrict__ cu) {
  int idx = blockIdx.x * blockDim.x + threadIdx.x;
  if (idx >= MM * AS) return;
  int a = idx % AS;
  float oma = 1.0f - alpha[a];
  int m = idx / AS;
#pragma unroll
  for (int n = 0; n < NN; ++n) {
    float s  = sigma[n * MM + m];
    float d  = delta_sig[n * MM + m] * 0.5f;
    float sg = s + d;                 // sig = sigma + delta_sig/2
    float dl = sg + d * oma;          // lower denominator
    float du = sg - d * oma;          // upper denominator
    cl[idx * NN + n] = -0.5f * LOG2E / (dl * dl);
    cu[idx * NN + n] = -0.5f * LOG2E / (du * du);
  }
}

// Issue one TDM copy: global (2D: 12928 x 1, f32) -> LDS at lds_off.
__device__ __forceinline__ void tdm_load_table(const float* gsrc,
                                               unsigned lds_off) {
  unsigned long long ga = (unsigned long long)(const void*)gsrc;
  u32x4 g0;
  g0.x = 1u;                                   // count=1, user mode
  g0.y = lds_off;                              // D#.lds_addr (byte offset)
  g0.z = (unsigned)(ga & 0xFFFFFFFFu);         // global_addr[31:0]
  g0.w = (unsigned)((ga >> 32) & 0x1FFFFFFu)   // global_addr[56:32]
         | (2u << 30);                         // type = 2 ("image")
  i32x8 g1;
  g1[0] = 0x00020000;                          // data_size=2 (4B), mask=0
  g1[1] = (int)((TBL_ELEMS & 0xFFFF) << 16);   // tensor_dim0[15:0]
  g1[2] = (int)(1u << 16);                     // tensor_dim0[31:16]=0, tensor_dim1=1
  g1[3] = (int)((TBL_ELEMS & 0xFFFF) << 16);   // tile_dim0 = 12928
  g1[4] = 1;                                   // tile_dim1 = 1, tile_dim2 = 0
  g1[5] = TBL_ELEMS;                           // tensor_dim0_stride[31:0]
  g1[6] = (int)((TBL_ELEMS & 0xFFFF) << 16);   // dim1_stride[15:0] (unused)
  g1[7] = 0;
  i32x4 gz4 = {};
  i32x8 gz8 = {};
  __builtin_amdgcn_tensor_load_to_lds(g0, g1, gz4, gz4, gz8, 0);
}

// Kernel 2: main. One wave handles 16 batch rows. Coefficient tables are
// TDM-staged into LDS once per block; for each alpha-tile (16 alphas) and
// each m, arg = d2(16x8) @ c(8x16) via two chained V_WMMA_F32_16X16X4_F32,
// then f = exp2(arg_lo)+exp2(arg_up), accumulate num/den, alpha-weighted
// combine, cross-lane reduce.
__global__ void __launch_bounds__(128) rbf_main(
    const float* __restrict__ X, const float* __restrict__ mu,
    const float* __restrict__ cl, const float* __restrict__ cu,
    const float* __restrict__ W, const float* __restrict__ bias,
    const float* __restrict__ alpha, float* __restrict__ out) {
  extern __shared__ float lds[];
  float* lds_cl = lds;                  // LDS byte offset 0
  float* lds_cu = lds + TBL_ELEMS;      // LDS byte offset TBL_BYTES

  const int lane = threadIdx.x & 31;
  const int wave = threadIdx.x >> 5;

  // --- Stage coefficient tables into LDS with the Tensor Data Mover.
  if (wave == 0) {
    tdm_load_table(cl, 0u);
    tdm_load_table(cu, (unsigned)TBL_BYTES);
    __builtin_amdgcn_s_wait_tensorcnt(0);
  }
  __syncthreads();

  const int b0   = blockIdx.x * 64 + wave * 16;
  const int row  = lane & 15;   // WMMA A: lane = batch row
  const int half = lane >> 4;   // selects K pair {0,1} vs {2,3}
  const int col  = lane & 15;   // WMMA B/D: lane = alpha column

  // A operands: d2[b][n] for all m, per ISA 16x4 f32 A layout.
  float2 xv0 = *(const float2*)(X + (b0 + row) * NN + 2 * half);
  float2 xv1 = *(const float2*)(X + (b0 + row) * NN + 4 + 2 * half);
  v2f A0[MM], A1[MM];
#pragma unroll
  for (int m = 0; m < MM; ++m) {
    int n0 = 2 * half;
    float t0 = xv0.x - mu[(n0    ) * MM + m];
    float t1 = xv0.y - mu[(n0 + 1) * MM + m];
    A0[m].x = t0 * t0; A0[m].y = t1 * t1;
    int n1 = 4 + 2 * half;
    float u0 = xv1.x - mu[(n1    ) * MM + m];
    float u1 = xv1.y - mu[(n1 + 1) * MM + m];
    A1[m].x = u0 * u0; A1[m].y = u1 * u1;
  }

  float w0[MM], w1[MM], w2[MM];
#pragma unroll
  for (int m = 0; m < MM; ++m) {
    w0[m] = W[m * PP + 0];
    w1[m] = W[m * PP + 1];
    w2[m] = W[m * PP + 2];
  }

  v8f y0 = {}, y1 = {}, y2 = {};
  float asum = 0.0f;

  for (int t = 0; t < 7; ++t) {          // 7 alpha tiles cover 101 alphas
    int a_idx = t * 16 + col;
    int a_c   = a_idx < AS ? a_idx : AS - 1;  // clamp B reads; weight 0 below
    v8f n0 = {}, n1 = {}, n2 = {}, den = {};
#pragma unroll
    for (int m = 0; m < MM; ++m) {
      const float* pl = lds_cl + (m * AS + a_c) * NN;
      const float* pu = lds_cu + (m * AS + a_c) * NN;
      // B operand (4x16 f32): lane = alpha col, VGPR pair holds K=2h, 2h+1
      v2f bl0 = *(const v2f*)(pl + 2 * half);
      v2f bl1 = *(const v2f*)(pl + 4 + 2 * half);
      v2f bu0 = *(const v2f*)(pu + 2 * half);
      v2f bu1 = *(const v2f*)(pu + 4 + 2 * half);
      v8f zero = {};
      v8f argl = __builtin_amdgcn_wmma_f32_16x16x4_f32(
          false, A0[m], false, bl0, (short)0, zero, false, false);
      argl = __builtin_amdgcn_wmma_f32_16x16x4_f32(
          false, A1[m], false, bl1, (short)0, argl, false, false);
      v8f argu = __builtin_amdgcn_wmma_f32_16x16x4_f32(
          false, A0[m], false, bu0, (short)0, zero, false, false);
      argu = __builtin_amdgcn_wmma_f32_16x16x4_f32(
          false, A1[m], false, bu1, (short)0, argu, false, false);
      v8f fs;
#pragma unroll
      for (int i = 0; i < 8; ++i)
        fs[i] = __builtin_amdgcn_exp2f(argl[i]) + __builtin_amdgcn_exp2f(argu[i]);
      den += fs;
      n0 += fs * w0[m];
      n1 += fs * w1[m];
      n2 += fs * w2[m];
    }
    float wa = (a_idx < AS) ? alpha[a_idx] : 0.0f;
    asum += wa;
#pragma unroll
    for (int i = 0; i < 8; ++i) {
      float r = wa * __builtin_amdgcn_rcpf(den[i]);
      y0[i] += n0[i] * r;
      y1[i] += n1[i] * r;
      y2[i] += n2[i] * r;
    }
  }

  // Reduce over alpha (16 lanes within each half); rows live in VGPR index.
#pragma unroll
  for (int s = 1; s <= 8; s <<= 1) {
    asum += __shfl_xor(asum, s, 32);
#pragma unroll
    for (int i = 0; i < 8; ++i) {
      y0[i] += __shfl_xor(y0[i], s, 32);
      y1[i] += __shfl_xor(y1[i], s, 32);
      y2[i] += __shfl_xor(y2[i], s, 32);
    }
  }

  if (col == 0) {  // lanes 0 (rows 0..7) and 16 (rows 8..15)
    float inv = 1.0f / asum;
    float bb0 = bias[0], bb1 = bias[1], bb2 = bias[2];
#pragma unroll
    for (int i = 0; i < 8; ++i) {
      int r = b0 + half * 8 + i;
      out[r * PP + 0] = y0[i] * inv + bb0;
      out[r * PP + 1] = y1[i] * inv + bb1;
      out[r * PP + 2] = y2[i] * inv + bb2;
    }
  }
}

extern "C" void kernel_launch(void* const* d_in, const int* in_sizes, int n_in,
                              void* d_out, int out_size, void* d_ws, size_t ws_size,
                              hipStream_t stream) {
  const float* X         = (const float*)d_in[0];
  const float* mu        = (const float*)d_in[1];
  const float* sigma     = (const float*)d_in[2];
  const float* delta_sig = (const float*)d_in[3];
  const float* W         = (const float*)d_in[4];
  const float* bias      = (const float*)d_in[5];
  const float* alpha     = (const float*)d_in[6];
  float* out = (float*)d_out;

  float* cl = (float*)d_ws;
  float* cu = cl + TBL_ELEMS;

  int B = in_sizes[0] / NN;

  build_tables<<<(MM * AS + 255) / 256, 256, 0, stream>>>(sigma, delta_sig, alpha, cl, cu);
  rbf_main<<<B / 64, 128, LDS_BYTES, stream>>>(X, mu, cl, cu, W, bias, alpha, out);
}